// PostProcess_4226247819682
// MI455X (gfx1250) — compile-verified
//
#include <hip/hip_runtime.h>
#include <math.h>

// ---------------------------------------------------------------------------
// DETR-style post-process for MI455X (gfx1250, wave32).
//   Kernel 1: softmax-max/argmax over 92 classes + box convert/scale.
//   Kernel 2: per-image soft-NMS scan, fully LDS-resident (64KB of 320KB),
//             staged in via gfx1250 async global->LDS DMA (ASYNCcnt path).
// ---------------------------------------------------------------------------

static constexpr int Bimg = 16;
static constexpr int Nq   = 2048;
static constexpr int Ccls = 92;
static constexpr int BN   = Bimg * Nq;

// ---- gfx1250 async global->LDS load (tracked with ASYNCcnt) ----------------
#if defined(__gfx1250__) &&                                                    \
    __has_builtin(__builtin_amdgcn_global_load_async_to_lds_b32) &&            \
    __has_builtin(__builtin_amdgcn_s_wait_asynccnt)
#define USE_ASYNC_LDS 1
#else
#define USE_ASYNC_LDS 0
#endif

__device__ __forceinline__ void g2l_b32(const float* g, float* l) {
#if USE_ASYNC_LDS
  __builtin_amdgcn_global_load_async_to_lds_b32(
      (__attribute__((address_space(1))) int*)(g),
      (__attribute__((address_space(3))) int*)(l),
      /*offset=*/0, /*cpol=*/0);
#else
  *l = *g;
#endif
}

__device__ __forceinline__ void async_wait_all() {
#if USE_ASYNC_LDS
  __builtin_amdgcn_s_wait_asynccnt(0);
#endif
}

// ---------------------------------------------------------------------------
// Kernel 1: one wave32 per (b,n) row of 92 logits.
//   m     = max over all 92 logits
//   denom = sum exp(l - m)
//   best  = max / argmax over first 91 logits (first-index tie-break)
//   score = exp(best - m) / denom          (== max softmax prob over :91)
// Also converts cxcywh -> xyxy and scales by (w,h,w,h) of target_sizes.
// ---------------------------------------------------------------------------
__global__ __launch_bounds__(256) void cls_box_kernel(
    const float* __restrict__ logits,    // [B,N,92]
    const float* __restrict__ boxes_in,  // [B,N,4]
    const float* __restrict__ tsizes,    // [B,2] = (h, w)
    float* __restrict__ out)             // [7*BN]
{
  const int wid  = threadIdx.x >> 5;
  const int lane = threadIdx.x & 31;
  const int row  = blockIdx.x * 8 + wid;      // 0 .. B*N-1 (exact)

  const float* lg = logits + (size_t)row * Ccls;

  const int   c2   = lane + 64;
  const bool  has2 = (c2 < Ccls);             // lane < 28
  float v0 = lg[lane];
  float v1 = lg[lane + 32];
  float v2 = has2 ? lg[c2] : -INFINITY;

  // wave max over all 92
  float m = fmaxf(v0, fmaxf(v1, v2));
  for (int s = 16; s; s >>= 1) m = fmaxf(m, __shfl_xor(m, s, 32));

  // wave sum of exp(l - m)
  float sum = __expf(v0 - m) + __expf(v1 - m) + (has2 ? __expf(v2 - m) : 0.f);
  for (int s = 16; s; s >>= 1) sum += __shfl_xor(sum, s, 32);

  // argmax over first 91 classes, first-index tie-break
  float bv = v0; int bi = lane;                 // lane in [0,32) < 91
  if (v1 > bv) { bv = v1; bi = lane + 32; }     // < 64 < 91
  if (c2 < Ccls - 1 && v2 > bv) { bv = v2; bi = c2; }
  for (int s = 16; s; s >>= 1) {
    float ov = __shfl_xor(bv, s, 32);
    int   oi = __shfl_xor(bi, s, 32);
    if (ov > bv || (ov == bv && oi < bi)) { bv = ov; bi = oi; }
  }

  if (lane == 0) {
    out[row]      = __expf(bv - m) / sum;   // score
    out[BN + row] = (float)bi;              // label (stored as f32)

    const int   b     = row / Nq;
    const float img_h = tsizes[b * 2 + 0];
    const float img_w = tsizes[b * 2 + 1];
    const float* bx = boxes_in + (size_t)row * 4;
    const float cx = bx[0], cy = bx[1], w = bx[2], h = bx[3];
    float* ob = out + 2 * BN + (size_t)row * 4;
    ob[0] = (cx - 0.5f * w) * img_w;
    ob[1] = (cy - 0.5f * h) * img_h;
    ob[2] = (cx + 0.5f * w) * img_w;
    ob[3] = (cy + 0.5f * h) * img_h;
  }
}

// ---------------------------------------------------------------------------
// Kernel 2: soft-NMS scan, one workgroup (256 thr / 8 waves) per image.
// Whole per-image state lives in LDS (64KB). 2047 sequential steps, each:
//   masked suffix argmax (wave32 shuffle reduce + 8-way LDS combine),
//   conditional row swap, IoU sweep + Gaussian decay of suffix scores.
// ---------------------------------------------------------------------------
__global__ __launch_bounds__(256) void soft_nms_kernel(
    float* __restrict__ out, const int* __restrict__ nms_flag)
{
  __shared__ float d0[Nq], d1[Nq], d2[Nq], d3[Nq];   // box columns
  __shared__ float idxc[Nq];                          // carried index column
  __shared__ float sc[Nq];                            // scores (mutated)
  __shared__ float area[Nq];
  __shared__ float lab[Nq];                           // labels, original order
  __shared__ float r_val[8];
  __shared__ int   r_idx[8];

  const int tid = threadIdx.x;
  const int b   = blockIdx.x;
  const int doNms = *nms_flag;

  const float* in_sc  = out + b * Nq;
  const float* in_lab = out + BN + b * Nq;
  const float* in_bx  = out + 2 * BN + (size_t)b * Nq * 4;

  // Stage everything into LDS via async DMA (deinterleave box columns).
  for (int j = tid; j < Nq; j += 256) {
    g2l_b32(in_sc + j,         &sc[j]);
    g2l_b32(in_lab + j,        &lab[j]);
    g2l_b32(in_bx + j * 4 + 0, &d0[j]);
    g2l_b32(in_bx + j * 4 + 1, &d1[j]);
    g2l_b32(in_bx + j * 4 + 2, &d2[j]);
    g2l_b32(in_bx + j * 4 + 3, &d3[j]);
  }
  async_wait_all();
  __syncthreads();

  for (int j = tid; j < Nq; j += 256) {
    idxc[j] = (float)j;
    area[j] = (d2[j] - d0[j] + 1.f) * (d3[j] - d1[j] + 1.f);
  }
  __syncthreads();

  if (doNms) {
    for (int i = 0; i < Nq - 1; ++i) {
      // ---- masked argmax over j in (i, N) ----
      float bv = -INFINITY; int bi = 0x7fffffff;
      for (int j = i + 1 + tid; j < Nq; j += 256) {
        float v = sc[j];
        if (v > bv) { bv = v; bi = j; }       // ascending scan => first max
      }
      for (int s = 16; s; s >>= 1) {
        float ov = __shfl_xor(bv, s, 32);
        int   oi = __shfl_xor(bi, s, 32);
        if (ov > bv || (ov == bv && oi < bi)) { bv = ov; bi = oi; }
      }
      if ((tid & 31) == 0) { r_val[tid >> 5] = bv; r_idx[tid >> 5] = bi; }
      __syncthreads();

      // ---- conditional swap of rows i and maxpos (thread 0) ----
      if (tid == 0) {
        float mv = r_val[0]; int mi = r_idx[0];
        for (int w = 1; w < 8; ++w) {
          float ov = r_val[w]; int oi = r_idx[w];
          if (ov > mv || (ov == mv && oi < mi)) { mv = ov; mi = oi; }
        }
        if (sc[i] < mv) {
          float t;
          t = d0[i];   d0[i]   = d0[mi];   d0[mi]   = t;
          t = d1[i];   d1[i]   = d1[mi];   d1[mi]   = t;
          t = d2[i];   d2[i]   = d2[mi];   d2[mi]   = t;
          t = d3[i];   d3[i]   = d3[mi];   d3[mi]   = t;
          t = idxc[i]; idxc[i] = idxc[mi]; idxc[mi] = t;
          t = sc[i];   sc[i]   = sc[mi];   sc[mi]   = t;
          t = area[i]; area[i] = area[mi]; area[mi] = t;
        }
      }
      __syncthreads();

      // ---- IoU sweep + Gaussian decay of suffix scores ----
      const float di0 = d0[i], di1 = d1[i], di2 = d2[i], di3 = d3[i];
      const float ai  = area[i];
      for (int j = i + 1 + tid; j < Nq; j += 256) {
        float yy1 = fmaxf(di0, d0[j]);
        float xx1 = fmaxf(di1, d1[j]);
        float yy2 = fminf(di2, d2[j]);
        float xx2 = fminf(di3, d3[j]);
        float w = fmaxf(0.f, xx2 - xx1 + 1.f);
        float h = fmaxf(0.f, yy2 - yy1 + 1.f);
        float inter = w * h;
        float ovr = inter / (ai + area[j] - inter);
        sc[j] *= __expf(-(ovr * ovr) * 2.0f);   // SIGMA = 0.5 => *2
      }
      __syncthreads();
    }
  }

  // ---- write results ----
  for (int j = tid; j < Nq; j += 256) {
    out[b * Nq + j]          = sc[j];
    int k = (int)idxc[j];
    out[BN + b * Nq + j]     = lab[k];                 // gathered labels
    float* ob = out + 2 * BN + (size_t)(b * Nq + j) * 4;
    ob[0] = d0[j]; ob[1] = d1[j]; ob[2] = d2[j]; ob[3] = d3[j];
    out[6 * BN + b * Nq + j] = idxc[j];                // keep indices
  }
}

// ---------------------------------------------------------------------------
extern "C" void kernel_launch(void* const* d_in, const int* in_sizes, int n_in,
                              void* d_out, int out_size, void* d_ws, size_t ws_size,
                              hipStream_t stream) {
  (void)in_sizes; (void)n_in; (void)d_ws; (void)ws_size; (void)out_size;
  const float* logits = (const float*)d_in[0];   // [16,2048,92]
  const float* boxes  = (const float*)d_in[1];   // [16,2048,4]
  const float* tsz    = (const float*)d_in[2];   // [16,2]
  const int*   nms    = (const int*)d_in[3];     // scalar flag
  float* out = (float*)d_out;                    // 7*BN floats

  cls_box_kernel<<<dim3(BN / 8), dim3(256), 0, stream>>>(logits, boxes, tsz, out);
  soft_nms_kernel<<<dim3(Bimg), dim3(256), 0, stream>>>(out, nms);
}